// Group_68805376082229
// MI455X (gfx1250) — compile-verified
//
#include <hip/hip_runtime.h>

// Problem constants (from reference): xyz (B=32, N=8192, 3) f32
// outputs: neighborhood (B, G=512, K=32, 3) then center (B, G, 3), concat flat.
#define BATCH 32
#define NPTS  8192
#define NGRP  512
#define KNN   32

typedef __attribute__((ext_vector_type(2))) float v2f;
typedef __attribute__((ext_vector_type(8))) float v8f;

// ---------------------------------------------------------------------------
// Kernel 1: farthest point sampling. One block per batch, 256 threads.
// Each thread owns 32 points (coords + running min-dist) entirely in VGPRs.
// Per step: update dists vs current centroid, argmax via shfl_xor reduction
// carrying (dist, idx, x, y, z) so the next centroid needs no memory fetch.
// Emits center coords (pre-update farthest, starting at point 0) — matches
// the jax.lax.scan convention in the reference.
// ---------------------------------------------------------------------------
__global__ __launch_bounds__(256) void fps_kernel(const float* __restrict__ xyz,
                                                  float* __restrict__ center)
{
    const int b   = blockIdx.x;
    const int tid = threadIdx.x;
    const float* px = xyz + (size_t)b * NPTS * 3;

    float xr[32], yr[32], zr[32], dmin[32];
#pragma unroll
    for (int j = 0; j < 32; ++j) {
        const int p = j * 256 + tid;
        xr[j] = px[p * 3 + 0];
        yr[j] = px[p * 3 + 1];
        zr[j] = px[p * 3 + 2];
        dmin[j] = 3.402823466e+38f;
    }

    __shared__ float rd[8], rx[8], ry[8], rz[8];
    __shared__ int   ri[8];
    __shared__ float cbuf[3];

    if (tid == 0) { cbuf[0] = xr[0]; cbuf[1] = yr[0]; cbuf[2] = zr[0]; }
    __syncthreads();

    const int wave = tid >> 5;
    const int lane = tid & 31;

    for (int g = 0; g < NGRP; ++g) {
        const float cx = cbuf[0], cy = cbuf[1], cz = cbuf[2];
        if (tid == 0) {
            float* co = center + ((size_t)b * NGRP + g) * 3;
            co[0] = cx; co[1] = cy; co[2] = cz;
        }

        float bd = -1.0f; int bi = 0;
        float bx = 0.f, by = 0.f, bz = 0.f;
#pragma unroll
        for (int j = 0; j < 32; ++j) {
            const float dx = xr[j] - cx, dy = yr[j] - cy, dz = zr[j] - cz;
            float d = dx * dx + dy * dy + dz * dz;
            d = fminf(dmin[j], d);
            dmin[j] = d;
            if (d > bd) {                 // ascending p within a thread: strict >
                bd = d; bi = j * 256 + tid;
                bx = xr[j]; by = yr[j]; bz = zr[j];
            }
        }

        // wave32 argmax reduction, tie -> lower index (jnp.argmax semantics)
#pragma unroll
        for (int off = 16; off; off >>= 1) {
            const float od = __shfl_xor(bd, off);
            const int   oi = __shfl_xor(bi, off);
            const float ox = __shfl_xor(bx, off);
            const float oy = __shfl_xor(by, off);
            const float oz = __shfl_xor(bz, off);
            if (od > bd || (od == bd && oi < bi)) {
                bd = od; bi = oi; bx = ox; by = oy; bz = oz;
            }
        }
        if (lane == 0) { rd[wave] = bd; ri[wave] = bi; rx[wave] = bx; ry[wave] = by; rz[wave] = bz; }
        __syncthreads();
        if (tid == 0) {
            float wd = rd[0]; int wi = ri[0];
            float wx = rx[0], wy = ry[0], wz = rz[0];
#pragma unroll
            for (int w = 1; w < 8; ++w) {
                if (rd[w] > wd || (rd[w] == wd && ri[w] < wi)) {
                    wd = rd[w]; wi = ri[w]; wx = rx[w]; wy = ry[w]; wz = rz[w];
                }
            }
            cbuf[0] = wx; cbuf[1] = wy; cbuf[2] = wz;
        }
        __syncthreads();
    }
}

// ---------------------------------------------------------------------------
// Kernel 2: 32-NN per center via V_WMMA_F32_16X16X4_F32.
// One wave per (batch, 16-center tile). Selection score for point m, center n:
//   s = ||x_m||^2 - 2 x_m . c_n   (drops the per-row constant ||c_n||^2,
//                                  which cannot change the top-k ordering)
// A (16 points x 4) per-lane fragment:  lanes 0-15: (x, y)   [K0,K1]
//                                       lanes 16-31: (z, ||x||^2) [K2,K3]
// B (4 x 16 centers) per-lane fragment: lanes 0-15: (-2cx, -2cy)
//                                       lanes 16-31: (-2cz, 1)
// D layout: lane l holds column n = l&15 with rows (l<16 ? 0..7 : 8..15).
// Two point-tiles (2 WMMAs) per loop iteration; a min-tree over all 16
// candidate scores feeds a single rarely-taken guard before the sorted
// register-resident top-32 insertion. Lane pairs (n, n+16) merge via LDS.
// ---------------------------------------------------------------------------
__global__ __launch_bounds__(32) void knn_group_kernel(const float* __restrict__ xyz,
                                                       const float* __restrict__ center,
                                                       float* __restrict__ nbh)
{
    const int b    = blockIdx.y;
    const int tile = blockIdx.x;
    const int lane = threadIdx.x;
    const int n    = lane & 15;
    const bool hi  = (lane >> 4) != 0;
    const int half = hi ? 1 : 0;

    const float* px = xyz + (size_t)b * NPTS * 3;
    const int g = tile * 16 + n;
    const float* cp = center + ((size_t)b * NGRP + g) * 3;
    const float cx = cp[0], cy = cp[1], cz = cp[2];

    v2f bf;                                  // branchless B-fragment setup
    bf[0] = -2.0f * (hi ? cz : cx);
    bf[1] = hi ? 1.0f : (-2.0f * cy);

    float bd[KNN];
    int   bi[KNN];
#pragma unroll
    for (int j = 0; j < KNN; ++j) { bd[j] = 3.402823466e+38f; bi[j] = 0; }

    for (int t = 0; t < NPTS / 32; ++t) {
        const int p0 = t * 32 + n;           // tile 2t
        const int p1 = p0 + 16;              // tile 2t+1
        const float* pa = px + (size_t)p0 * 3;
        const float* pb = px + (size_t)p1 * 3;
        if (t + 4 < NPTS / 32)
            __builtin_prefetch(px + (size_t)((t + 4) * 32 + n) * 3, 0, 3);

        const float x0 = pa[0], y0 = pa[1], z0 = pa[2];
        const float x1 = pb[0], y1 = pb[1], z1 = pb[2];
        const float n0 = x0 * x0 + y0 * y0 + z0 * z0;
        const float n1 = x1 * x1 + y1 * y1 + z1 * z1;

        v2f af0, af1;                        // branchless A-fragment setup
        af0[0] = hi ? z0 : x0;  af0[1] = hi ? n0 : y0;
        af1[0] = hi ? z1 : x1;  af1[1] = hi ? n1 : y1;

        v8f acc0 = {}, acc1 = {};
        acc0 = __builtin_amdgcn_wmma_f32_16x16x4_f32(
            false, af0, false, bf, (short)0, acc0, false, false);
        acc1 = __builtin_amdgcn_wmma_f32_16x16x4_f32(
            false, af1, false, bf, (short)0, acc1, false, false);

        // min-tree over the 16 candidate scores: one branch, almost never taken
        float m = fminf(acc0[0], acc0[1]);
        m = fminf(m, fminf(acc0[2], acc0[3]));
        m = fminf(m, fminf(acc0[4], acc0[5]));
        m = fminf(m, fminf(acc0[6], acc0[7]));
        m = fminf(m, fminf(acc1[0], acc1[1]));
        m = fminf(m, fminf(acc1[2], acc1[3]));
        m = fminf(m, fminf(acc1[4], acc1[5]));
        m = fminf(m, fminf(acc1[6], acc1[7]));

        if (m < bd[KNN - 1]) {
            const int base0 = t * 32 + half * 8;
            const int base1 = base0 + 16;
#pragma unroll
            for (int r = 0; r < 8; ++r) {
                const float v = acc0[r];
                if (v < bd[KNN - 1]) {       // guarded insert into sorted top-32
                    bd[KNN - 1] = v; bi[KNN - 1] = base0 + r;
#pragma unroll
                    for (int j = KNN - 1; j > 0; --j) {
                        if (bd[j] < bd[j - 1]) {
                            const float td = bd[j]; bd[j] = bd[j - 1]; bd[j - 1] = td;
                            const int   ti = bi[j]; bi[j] = bi[j - 1]; bi[j - 1] = ti;
                        }
                    }
                }
            }
#pragma unroll
            for (int r = 0; r < 8; ++r) {
                const float v = acc1[r];
                if (v < bd[KNN - 1]) {
                    bd[KNN - 1] = v; bi[KNN - 1] = base1 + r;
#pragma unroll
                    for (int j = KNN - 1; j > 0; --j) {
                        if (bd[j] < bd[j - 1]) {
                            const float td = bd[j]; bd[j] = bd[j - 1]; bd[j - 1] = td;
                            const int   ti = bi[j]; bi[j] = bi[j - 1]; bi[j - 1] = ti;
                        }
                    }
                }
            }
        }
    }

    // merge the two half-row lists per center through LDS
    __shared__ float sd[32][KNN];
    __shared__ int   si[32][KNN];
#pragma unroll
    for (int j = 0; j < KNN; ++j) { sd[lane][j] = bd[j]; si[lane][j] = bi[j]; }
    __syncthreads();

    if (!hi) {
        float* outp = nbh + ((size_t)b * NGRP + g) * KNN * 3;
        int ia = 0, ib = 0;
        for (int k = 0; k < KNN; ++k) {
            const float da = sd[n][ia],      db = sd[n + 16][ib];
            const int   ja = si[n][ia],      jb = si[n + 16][ib];
            const bool takeA = (da < db) || (da == db && ja < jb);
            const int pi = takeA ? ja : jb;
            if (takeA) ++ia; else ++ib;
            const float* pp = px + (size_t)pi * 3;
            outp[k * 3 + 0] = pp[0] - cx;
            outp[k * 3 + 1] = pp[1] - cy;
            outp[k * 3 + 2] = pp[2] - cz;
        }
    }
}

// ---------------------------------------------------------------------------
extern "C" void kernel_launch(void* const* d_in, const int* in_sizes, int n_in,
                              void* d_out, int out_size, void* d_ws, size_t ws_size,
                              hipStream_t stream)
{
    (void)in_sizes; (void)n_in; (void)out_size; (void)d_ws; (void)ws_size;
    const float* xyz = (const float*)d_in[0];
    float* out    = (float*)d_out;
    float* nbh    = out;                                          // B*G*K*3
    float* center = out + (size_t)BATCH * NGRP * KNN * 3;         // B*G*3

    fps_kernel<<<BATCH, 256, 0, stream>>>(xyz, center);
    knn_group_kernel<<<dim3(NGRP / 16, BATCH), 32, 0, stream>>>(xyz, center, nbh);
}